// Deformer_53807350284832
// MI455X (gfx1250) — compile-verified
//
#include <hip/hip_runtime.h>
#include <hip/hip_bf16.h>

typedef __attribute__((ext_vector_type(2))) float v2f;
typedef __attribute__((ext_vector_type(8))) float v8f;

#define BB 64
#define HH 256
#define WW 256
#define CC 3
#define KK 20

// ---------------------------------------------------------------------------
// Affine warp of one output pixel (bilinear, zero outside), matches reference.
// ---------------------------------------------------------------------------
__device__ __forceinline__ void warp_pixel(const float* __restrict__ img_b,
                                           float a0, float a1, float a2,
                                           float b0, float b1, float b2,
                                           int x, int y,
                                           float& outr, float& outg, float& outb) {
    float xf = (float)x, yf = (float)y;
    float xs = a0 * xf + a1 * yf + a2;
    float ys = b0 * xf + b1 * yf + b2;
    float x0f = floorf(xs);
    float y0f = floorf(ys);
    float wx = xs - x0f;
    float wy = ys - y0f;
    int xi = (int)x0f;
    int yi = (int)y0f;

    float r00, g00, b00, r01, g01, b01, r10, g10, b10, r11, g11, b11;
    auto corner = [&](int yy, int xx, float& cr, float& cg, float& cb) {
        bool valid = (yy >= 0) & (yy < HH) & (xx >= 0) & (xx < WW);
        int yc = min(max(yy, 0), HH - 1);
        int xc = min(max(xx, 0), WW - 1);
        const float* p = img_b + (yc * WW + xc) * CC;
        float m = valid ? 1.0f : 0.0f;
        cr = p[0] * m; cg = p[1] * m; cb = p[2] * m;
    };
    corner(yi,     xi,     r00, g00, b00);
    corner(yi,     xi + 1, r01, g01, b01);
    corner(yi + 1, xi,     r10, g10, b10);
    corner(yi + 1, xi + 1, r11, g11, b11);

    float omx = 1.0f - wx, omy = 1.0f - wy;
    outr = omy * (omx * r00 + wx * r01) + wy * (omx * r10 + wx * r11);
    outg = omy * (omx * g00 + wx * g01) + wy * (omx * g10 + wx * g11);
    outb = omy * (omx * b00 + wx * b01) + wy * (omx * b10 + wx * b11);
}

__device__ __forceinline__ void make_affine(const float* zoom, const float* rot,
                                            const float* shy, const float* ty,
                                            const float* tx, int b,
                                            float& a0, float& a1, float& a2,
                                            float& b0, float& b1, float& b2) {
    const float cy = (HH - 1) * 0.5f;
    const float cx = (WW - 1) * 0.5f;
    float z = zoom[b], r = rot[b], s = shy[b];
    float si, co;
    __sincosf(r, &si, &co);
    a0 = z * co;
    a1 = -z * (si + s);
    b0 = z * si;
    b1 = z * co;
    a2 = -a0 * cx - a1 * cy + cx + tx[b];
    b2 = -b0 * cx - b1 * cy + cy + ty[b];
}

// ---------------------------------------------------------------------------
// Deterministic block (256 thr = 8 waves) reduction of 3 channel sums.
// Wave level: wave32 shfl_xor tree. Cross-wave: two chained
// V_WMMA_F32_16X16X4_F32 ops. A-matrix rows 0..2 hold the per-wave partials
// for channels R,G,B in the K slots (lane l -> K0,K1 ; lane l+16 -> K2,K3 per
// ISA 16x4 A layout), B = all-ones, so D[c][0] = total channel sum.
// ---------------------------------------------------------------------------
__device__ __forceinline__ void block_reduce3_store(float sr, float sg, float sb,
                                                    float* __restrict__ dst3) {
    #pragma unroll
    for (int off = 16; off >= 1; off >>= 1) {
        sr += __shfl_xor(sr, off, 32);
        sg += __shfl_xor(sg, off, 32);
        sb += __shfl_xor(sb, off, 32);
    }
    __shared__ float s_part[8][3];
    int lane = threadIdx.x & 31;
    int wid  = threadIdx.x >> 5;
    if (lane == 0) {
        s_part[wid][0] = sr; s_part[wid][1] = sg; s_part[wid][2] = sb;
    }
    __syncthreads();
    if (threadIdx.x < 32) {               // wave 0, EXEC all ones at the WMMAs
        int l = threadIdx.x;
        float a00 = 0.f, a01 = 0.f, a10 = 0.f, a11 = 0.f;
        if (l < 3) {                       // A rows 0..2, K0/K1 (waves 0,1) + second wmma waves 4,5
            a00 = s_part[0][l]; a01 = s_part[1][l];
            a10 = s_part[4][l]; a11 = s_part[5][l];
        } else if (l >= 16 && l < 19) {    // same rows, K2/K3 (waves 2,3) + waves 6,7
            int c = l - 16;
            a00 = s_part[2][c]; a01 = s_part[3][c];
            a10 = s_part[6][c]; a11 = s_part[7][c];
        }
        v2f A1 = {a00, a01};
        v2f A2 = {a10, a11};
        v2f Bones = {1.0f, 1.0f};
        v8f D = {};
        D = __builtin_amdgcn_wmma_f32_16x16x4_f32(false, A1, false, Bones,
                                                  (short)0, D, false, false);
        D = __builtin_amdgcn_wmma_f32_16x16x4_f32(false, A2, false, Bones,
                                                  (short)0, D, false, false);
        if (l == 0) {                      // lane 0 holds D[0][0],D[1][0],D[2][0]
            dst3[0] = D[0]; dst3[1] = D[1]; dst3[2] = D[2];
        }
    }
}

// ---------------------------------------------------------------------------
// Pass 1: warp + per-block channel sums (one 256-pixel row per block).
// ---------------------------------------------------------------------------
__global__ void warp_sum_kernel(const float* __restrict__ images,
                                const float* __restrict__ zoom,
                                const float* __restrict__ rot,
                                const float* __restrict__ shy,
                                const float* __restrict__ ty,
                                const float* __restrict__ tx,
                                float* __restrict__ partials) {
    int p = blockIdx.x * 256 + threadIdx.x;
    int b = p >> 16;
    int y = (p >> 8) & 255;
    int x = p & 255;
    float a0, a1, a2, b0, b1, b2;
    make_affine(zoom, rot, shy, ty, tx, b, a0, a1, a2, b0, b1, b2);
    const float* img_b = images + (size_t)b * HH * WW * CC;
    float r, g, bl;
    warp_pixel(img_b, a0, a1, a2, b0, b1, b2, x, y, r, g, bl);
    block_reduce3_store(r, g, bl, partials + (size_t)blockIdx.x * 3);
}

// ---------------------------------------------------------------------------
// Pass 1b: fold 256 block-partials per image -> per-image channel sums.
// ---------------------------------------------------------------------------
__global__ void reduce_kernel(const float* __restrict__ partials,
                              float* __restrict__ sums) {
    int b = blockIdx.x;
    const float* p = partials + ((size_t)b * 256 + threadIdx.x) * 3;
    block_reduce3_store(p[0], p[1], p[2], sums + (size_t)b * 3);
}

// ---------------------------------------------------------------------------
// Pass 2: recompute warp (gathers hit L2; whole input fits in 192MB L2),
// contrast about the mean, brightness, cutouts, clip, store.
// ---------------------------------------------------------------------------
__global__ void finalize_kernel(const float* __restrict__ images,
                                const float* __restrict__ zoom,
                                const float* __restrict__ rot,
                                const float* __restrict__ shy,
                                const float* __restrict__ ty,
                                const float* __restrict__ tx,
                                const float* __restrict__ contrast,
                                const float* __restrict__ brightness,
                                const float* __restrict__ fill_is_one,
                                const int* __restrict__ hs,
                                const int* __restrict__ wsz,
                                const int* __restrict__ y0a,
                                const int* __restrict__ x0a,
                                const float* __restrict__ sums,
                                float* __restrict__ out) {
    int p = blockIdx.x * 256 + threadIdx.x;
    int b = p >> 16;           // block-uniform
    int y = (p >> 8) & 255;
    int x = p & 255;

    // Stage cutout rects for this image into LDS (b is uniform per block).
    __shared__ int   s_y0[KK], s_y1[KK], s_x0[KK], s_x1[KK];
    __shared__ float s_fill[KK];
    if (threadIdx.x < KK) {
        int k = threadIdx.x;
        int hv = hs[b * KK + k];
        int wv = wsz[b * KK + k];
        int yy = min(y0a[b * KK + k], HH - hv);
        int xx = min(x0a[b * KK + k], WW - wv);
        s_y0[k] = yy; s_y1[k] = yy + hv;
        s_x0[k] = xx; s_x1[k] = xx + wv;
        s_fill[k] = fill_is_one[b * KK + k];
    }
    __syncthreads();

    float a0, a1, a2, b0, b1, b2;
    make_affine(zoom, rot, shy, ty, tx, b, a0, a1, a2, b0, b1, b2);
    const float* img_b = images + (size_t)b * HH * WW * CC;
    float r, g, bl;
    warp_pixel(img_b, a0, a1, a2, b0, b1, b2, x, y, r, g, bl);

    const float inv_n = 1.0f / (float)(HH * WW);
    float mr = sums[b * 3 + 0] * inv_n;
    float mg = sums[b * 3 + 1] * inv_n;
    float mb = sums[b * 3 + 2] * inv_n;
    float ct = contrast[b];
    float br = brightness[b];
    r = (r - mr) * ct + mr + br;
    g = (g - mg) * ct + mg + br;
    bl = (bl - mb) * ct + mb + br;

    bool er = false, bo = false;
    #pragma unroll
    for (int k = 0; k < KK; ++k) {
        bool in = (y >= s_y0[k]) & (y < s_y1[k]) & (x >= s_x0[k]) & (x < s_x1[k]);
        if (in) {
            if (s_fill[k] > 0.5f) bo = true; else er = true;
        }
    }
    // erase -> 0, then blot -> 1 (blot wins, matching reference algebra)
    r = er ? 0.0f : r;  g = er ? 0.0f : g;  bl = er ? 0.0f : bl;
    r = bo ? 1.0f : r;  g = bo ? 1.0f : g;  bl = bo ? 1.0f : bl;
    r  = fminf(fmaxf(r, 0.0f), 1.0f);
    g  = fminf(fmaxf(g, 0.0f), 1.0f);
    bl = fminf(fmaxf(bl, 0.0f), 1.0f);

    float* o = out + (size_t)p * 3;
    o[0] = r; o[1] = g; o[2] = bl;
}

// ---------------------------------------------------------------------------
extern "C" void kernel_launch(void* const* d_in, const int* in_sizes, int n_in,
                              void* d_out, int out_size, void* d_ws, size_t ws_size,
                              hipStream_t stream) {
    const float* images     = (const float*)d_in[0];
    const float* zoom       = (const float*)d_in[1];
    const float* rot        = (const float*)d_in[2];
    const float* shy        = (const float*)d_in[3];
    const float* ty         = (const float*)d_in[4];
    const float* tx         = (const float*)d_in[5];
    const float* contrast   = (const float*)d_in[6];
    const float* brightness = (const float*)d_in[7];
    const float* fill       = (const float*)d_in[8];
    const int*   hs         = (const int*)d_in[9];
    const int*   wsz        = (const int*)d_in[10];
    const int*   y0a        = (const int*)d_in[11];
    const int*   x0a        = (const int*)d_in[12];
    float* out = (float*)d_out;

    const int nblocks = BB * HH * WW / 256;       // 16384
    float* partials = (float*)d_ws;               // nblocks*3 floats
    float* sums     = partials + (size_t)nblocks * 3;  // BB*3 floats

    warp_sum_kernel<<<nblocks, 256, 0, stream>>>(images, zoom, rot, shy, ty, tx,
                                                 partials);
    reduce_kernel<<<BB, 256, 0, stream>>>(partials, sums);
    finalize_kernel<<<nblocks, 256, 0, stream>>>(images, zoom, rot, shy, ty, tx,
                                                 contrast, brightness, fill,
                                                 hs, wsz, y0a, x0a, sums, out);
}